// cross_attention_64604898067012
// MI455X (gfx1250) — compile-verified
//
#include <hip/hip_runtime.h>
#include <math.h>

// ---------------- problem constants (match reference) ----------------
#define kBS    8
#define kNQ    4096
#define kC     256
#define kHeads 8
#define kPts   4
#define kHD    32
#define kHV    23
#define kWV    46
#define kHW    (kHV * kWV)     // 1058
#define kNV    8
#define kNPack 112             // 64 (Wo) + 32 (Wa) + 1 (Wp) + 15 pad
#define kMQ    (kBS * kNQ)     // 32768
#define kMV    (kBS * kHW)     // 8464 (= 529*16, tile-exact)

typedef float v2f __attribute__((ext_vector_type(2)));
typedef float v8f __attribute__((ext_vector_type(8)));

// ---------------- pack W[K,N] -> paired layout for b64 B-fragment loads --
// Wpair[(p*N + n)*2 + j] = W[2p + j][n],  p in [0, K/2)
__global__ void msda_pack_pairs_kernel(const float* __restrict__ W,
                                       float* __restrict__ Wpair, int N, int K) {
  int total = (K / 2) * N;
  for (int idx = blockIdx.x * blockDim.x + threadIdx.x; idx < total;
       idx += gridDim.x * blockDim.x) {
    int p = idx / N, n = idx - p * N;
    Wpair[(size_t)idx * 2 + 0] = W[(size_t)(2 * p + 0) * N + n];
    Wpair[(size_t)idx * 2 + 1] = W[(size_t)(2 * p + 1) * N + n];
  }
}

// ---------------- K0: build paired [Wo|Wa|Wp|0], u = Wout@Wp, c0 = bout.Wp
__global__ void msda_prep_kernel(const float* __restrict__ Wo, const float* __restrict__ bo,
                                 const float* __restrict__ Wa, const float* __restrict__ ba,
                                 const float* __restrict__ Wout, const float* __restrict__ bout,
                                 const float* __restrict__ Wp, const float* __restrict__ bp,
                                 float* __restrict__ WpackP, float* __restrict__ bpack,
                                 float* __restrict__ u, float* __restrict__ c0) {
  int t = threadIdx.x;  // 0..255
  float s = 0.f;
  for (int j = 0; j < kC; ++j) s += Wout[t * kC + j] * Wp[j];
  u[t] = s;
  if (t == 0) {
    float cc = 0.f;
    for (int j = 0; j < kC; ++j) cc += bout[j] * Wp[j];
    c0[0] = cc;
  }
  // paired pack: WpackP[(p*112 + n)*2 + j] = packrow(2p+j, n), p in [0,128)
  for (int idx = t; idx < (kC / 2) * kNPack; idx += 256) {
    int p = idx / kNPack, n = idx - p * kNPack;
#pragma unroll
    for (int j = 0; j < 2; ++j) {
      int r = 2 * p + j;
      float v;
      if (n < 64)       v = Wo[r * 64 + n];
      else if (n < 96)  v = Wa[r * 32 + (n - 64)];
      else if (n == 96) v = Wp[r];
      else              v = 0.f;
      WpackP[(size_t)idx * 2 + j] = v;
    }
  }
  if (t < kNPack) {
    float v;
    if (t < 64)       v = bo[t];
    else if (t < 96)  v = ba[t - 64];
    else if (t == 96) v = bp[0];
    else              v = 0.f;
    bpack[t] = v;
  }
}

// ---------------- f32 WMMA GEMM: out = A@W + bias (+ addscale*add) -------
// One wave computes a 16 x (16*NT) strip: per k-step, one A b64 load is
// shared by NT V_WMMA_F32_16X16X4_F32 ops (independent accumulators, so no
// D->A/B WMMA hazards between them). W is in paired layout (see above).
template <int NT>
__global__ void msda_wmma_gemm_f32(const float* __restrict__ A, const float* __restrict__ Wpair,
                                   const float* __restrict__ bias, const float* __restrict__ add,
                                   float addscale, float* __restrict__ out,
                                   int M, int N, int K) {
  int waveId = blockIdx.x * (blockDim.x >> 5) + (threadIdx.x >> 5);
  int lane = threadIdx.x & 31;
  int nGroupsN = N / (16 * NT);
  int nWork = (M >> 4) * nGroupsN;
  if (waveId >= nWork) return;                // wave-uniform: EXEC stays all-ones
  int mt = waveId / nGroupsN, ng = waveId - mt * nGroupsN;
  int row0 = mt << 4, col0 = ng * (16 * NT);
  int half = lane >> 4;                       // K-pair select (ISA A/B 16x4 layout)
  int r    = lane & 15;                       // A row / B col within tile
  v8f c[NT];
#pragma unroll
  for (int t = 0; t < NT; ++t) c[t] = (v8f){};
  const float* arow = A + (size_t)(row0 + r) * K + 2 * half;
  for (int k = 0; k < K; k += 4) {
    v2f a = *(const v2f*)(arow + k);          // A[row][k+2h .. k+2h+1]
    const float* wb = Wpair + ((size_t)(k / 2 + half) * N + col0 + r) * 2;
#pragma unroll
    for (int t = 0; t < NT; ++t) {
      v2f b = *(const v2f*)(wb + t * 32);     // W[k+2h..k+2h+1][col0+16t+r]
      c[t] = __builtin_amdgcn_wmma_f32_16x16x4_f32(false, a, false, b, (short)0,
                                                   c[t], false, false);
    }
  }
#pragma unroll
  for (int t = 0; t < NT; ++t) {
    int ccol = col0 + t * 16 + r;
    float bcol = bias ? bias[ccol] : 0.f;
#pragma unroll
    for (int i = 0; i < 8; ++i) {
      int rr = row0 + i + 8 * half;           // C/D layout: VGPR i -> M = i + 8*half
      float v = c[t][i] + bcol;
      if (add) v += addscale * add[(size_t)rr * N + ccol];
      out[(size_t)rr * N + ccol] = v;
    }
  }
}

// ---------------- K2b: per-head softmax over PTS ----------------
__global__ void msda_attn_softmax_kernel(const float* __restrict__ proj,
                                         float* __restrict__ attn) {
  int tid = blockIdx.x * blockDim.x + threadIdx.x;
  if (tid >= kMQ) return;
  const float* row = proj + (size_t)tid * kNPack + 64;
#pragma unroll
  for (int h = 0; h < kHeads; ++h) {
    float l0 = row[h * 4 + 0], l1 = row[h * 4 + 1];
    float l2 = row[h * 4 + 2], l3 = row[h * 4 + 3];
    float m = fmaxf(fmaxf(l0, l1), fmaxf(l2, l3));
    float e0 = __expf(l0 - m), e1 = __expf(l1 - m);
    float e2 = __expf(l2 - m), e3 = __expf(l3 - m);
    float inv = 1.f / (e0 + e1 + e2 + e3);
    float* o = attn + (size_t)tid * 32 + h * 4;
    o[0] = e0 * inv; o[1] = e1 * inv; o[2] = e2 * inv; o[3] = e3 * inv;
  }
}

// ---------------- bilinear fetch (zero padding, per-head channel) ------
__device__ __forceinline__ float msda_fetch(const float* __restrict__ vb,
                                            int yi, int xi, int ch) {
  if (xi < 0 || xi >= kWV || yi < 0 || yi >= kHV) return 0.f;
  return vb[((size_t)yi * kWV + xi) * kC + ch];
}

__device__ __forceinline__ float msda_bilin(const float* __restrict__ vb,
                                            int ch, float x, float y) {
  float x0f = floorf(x), y0f = floorf(y);
  int x0 = (int)x0f, y0 = (int)y0f;
  float wx1 = x - x0f, wy1 = y - y0f;
  float wx0 = 1.f - wx1, wy0 = 1.f - wy1;
  float g00 = msda_fetch(vb, y0,     x0,     ch);
  float g10 = msda_fetch(vb, y0,     x0 + 1, ch);
  float g01 = msda_fetch(vb, y0 + 1, x0,     ch);
  float g11 = msda_fetch(vb, y0 + 1, x0 + 1, ch);
  return g00 * (wx0 * wy0) + g10 * (wx1 * wy0) + g01 * (wx0 * wy1) + g11 * (wx1 * wy1);
}

// ---------------- K3: pass A — height logits p[v] = s_v.u + q.Wp + bout.Wp + bp
// one wave per (b,nq); lane = head-dim channel; deterministic (no atomics).
__global__ void msda_height_logits_kernel(const float* __restrict__ v2d,
                                          const float* __restrict__ proj,
                                          const float* __restrict__ attn,
                                          const float* __restrict__ grd_col,
                                          const float* __restrict__ grd_row,
                                          const float* __restrict__ u,
                                          const float* __restrict__ c0,
                                          float* __restrict__ p) {
  int wave = blockIdx.x * (blockDim.x >> 5) + (threadIdx.x >> 5);
  int lane = threadIdx.x & 31;
  if (wave >= kMQ) return;
  int bq = wave;
  int nq = bq & (kNQ - 1);
  int b  = bq >> 12;
  const float* vb = v2d + (size_t)b * kHW * kC;
  float base = proj[(size_t)bq * kNPack + 96] + c0[0];  // q.Wp + bp + bout.Wp
  float pv[kNV];
#pragma unroll
  for (int v = 0; v < kNV; ++v) pv[v] = 0.f;
  for (int h = 0; h < kHeads; ++h) {
    const float* offp = proj + (size_t)bq * kNPack + h * 8;
    const float* ap   = attn + (size_t)bq * 32 + h * 4;
    float ox[kPts], oy[kPts], aw[kPts];
#pragma unroll
    for (int pt = 0; pt < kPts; ++pt) {
      ox[pt] = offp[pt * 2]; oy[pt] = offp[pt * 2 + 1]; aw[pt] = ap[pt];
    }
    int ch = h * kHD + lane;
    float uv = u[ch];
#pragma unroll
    for (int v = 0; v < kNV; ++v) {
      float rx = grd_col[(size_t)nq * kNV + v] * kWV - 0.5f;
      float ry = grd_row[(size_t)nq * kNV + v] * kHV - 0.5f;
      float sval = 0.f;
#pragma unroll
      for (int pt = 0; pt < kPts; ++pt)
        sval += aw[pt] * msda_bilin(vb, ch, rx + ox[pt], ry + oy[pt]);
      pv[v] += sval * uv;
    }
  }
#pragma unroll
  for (int v = 0; v < kNV; ++v) {
    float t = pv[v];
#pragma unroll
    for (int m = 16; m >= 1; m >>= 1) t += __shfl_xor(t, m, 32);
    if (lane == 0) p[(size_t)v * kMQ + bq] = t + base;
  }
}

// ---------------- K4a: softmax over NV (in place) + argmax index --------
__global__ void msda_height_softmax_kernel(float* __restrict__ p,
                                           float* __restrict__ out_idx) {
  int tid = blockIdx.x * blockDim.x + threadIdx.x;
  if (tid >= kMQ) return;
  float vals[kNV];
  float m = -3.4e38f;
  int idx = 0;
#pragma unroll
  for (int v = 0; v < kNV; ++v) {
    vals[v] = p[(size_t)v * kMQ + tid];
    if (vals[v] > m) { m = vals[v]; idx = v; }   // first max, like jnp.argmax
  }
  float sum = 0.f;
#pragma unroll
  for (int v = 0; v < kNV; ++v) { vals[v] = __expf(vals[v] - m); sum += vals[v]; }
  float inv = 1.f / sum;
#pragma unroll
  for (int v = 0; v < kNV; ++v) p[(size_t)v * kMQ + tid] = vals[v] * inv;
  out_idx[tid] = (float)idx;
}

// ---------------- K4b: pass B — acc = sum_v w_v * s_v ------------------
__global__ void msda_weighted_sample_kernel(const float* __restrict__ v2d,
                                            const float* __restrict__ proj,
                                            const float* __restrict__ attn,
                                            const float* __restrict__ grd_col,
                                            const float* __restrict__ grd_row,
                                            const float* __restrict__ w,
                                            float* __restrict__ acc) {
  int wave = blockIdx.x * (blockDim.x >> 5) + (threadIdx.x >> 5);
  int lane = threadIdx.x & 31;
  if (wave >= kMQ) return;
  int bq = wave;
  int nq = bq & (kNQ - 1);
  int b  = bq >> 12;
  const float* vb = v2d + (size_t)b * kHW * kC;
  float wv[kNV];
#pragma unroll
  for (int v = 0; v < kNV; ++v) wv[v] = w[(size_t)v * kMQ + bq];
  for (int h = 0; h < kHeads; ++h) {
    const float* offp = proj + (size_t)bq * kNPack + h * 8;
    const float* ap   = attn + (size_t)bq * 32 + h * 4;
    float ox[kPts], oy[kPts], aw[kPts];
#pragma unroll
    for (int pt = 0; pt < kPts; ++pt) {
      ox[pt] = offp[pt * 2]; oy[pt] = offp[pt * 2 + 1]; aw[pt] = ap[pt];
    }
    int ch = h * kHD + lane;
    float accv = 0.f;
#pragma unroll
    for (int v = 0; v < kNV; ++v) {
      float rx = grd_col[(size_t)nq * kNV + v] * kWV - 0.5f;
      float ry = grd_row[(size_t)nq * kNV + v] * kHV - 0.5f;
      float sval = 0.f;
#pragma unroll
      for (int pt = 0; pt < kPts; ++pt)
        sval += aw[pt] * msda_bilin(vb, ch, rx + ox[pt], ry + oy[pt]);
      accv += wv[v] * sval;
    }
    acc[(size_t)bq * kC + ch] = accv;  // coalesced 32-float store per head
  }
}

// ---------------- host launcher ----------------------------------------
extern "C" void kernel_launch(void* const* d_in, const int* in_sizes, int n_in,
                              void* d_out, int out_size, void* d_ws, size_t ws_size,
                              hipStream_t stream) {
  (void)in_sizes; (void)n_in; (void)out_size; (void)ws_size;
  const float* query   = (const float*)d_in[0];
  const float* value   = (const float*)d_in[1];
  const float* grd_col = (const float*)d_in[2];
  const float* grd_row = (const float*)d_in[3];
  const float* Wv   = (const float*)d_in[4];
  const float* bv   = (const float*)d_in[5];
  const float* Wo   = (const float*)d_in[6];
  const float* bo   = (const float*)d_in[7];
  const float* Wa   = (const float*)d_in[8];
  const float* ba   = (const float*)d_in[9];
  const float* Wout = (const float*)d_in[10];
  const float* bout = (const float*)d_in[11];
  const float* Wp   = (const float*)d_in[12];
  const float* bp   = (const float*)d_in[13];

  float* out     = (float*)d_out;                       // [BS*NQ, C]
  float* out_idx = out + (size_t)kMQ * kC;              // [BS*NQ] argmax indices

  // workspace layout (floats), all offsets 16B-aligned
  float* ws     = (float*)d_ws;
  float* v2d    = ws;                                   // 2,166,784
  float* WvP    = v2d    + (size_t)kMV * kC;            //    65,536
  float* WoutP  = WvP    + (size_t)kC * kC;             //    65,536
  float* WpackP = WoutP  + (size_t)kC * kC;             //    28,672
  float* bpack  = WpackP + (size_t)kC * kNPack;         //       128
  float* u      = bpack + 128;                          //       256
  float* c0     = u + kC;                               //        16
  float* proj   = c0 + 16;                              // 3,670,016
  float* attn   = proj + (size_t)kMQ * kNPack;          // 1,048,576
  float* p      = attn + (size_t)kMQ * 32;              //   262,144
  float* acc    = p + (size_t)kNV * kMQ;                // 8,388,608
  // total ~62.8 MB

  // K0a/K0b: pack Wv / Wout into paired layout; K0c: prep (WpackP, u, c0, bpack)
  msda_pack_pairs_kernel<<<dim3(128), dim3(256), 0, stream>>>(Wv, WvP, kC, kC);
  msda_pack_pairs_kernel<<<dim3(128), dim3(256), 0, stream>>>(Wout, WoutP, kC, kC);
  msda_prep_kernel<<<dim3(1), dim3(256), 0, stream>>>(
      Wo, bo, Wa, ba, Wout, bout, Wp, bp, WpackP, bpack, u, c0);

  // K1: v2d = value @ Wv + bv        (M=8464, N=256) NT=4 -> 529*4 waves
  {
    int waves = (kMV / 16) * (kC / 64);
    msda_wmma_gemm_f32<4><<<dim3((waves + 7) / 8), dim3(256), 0, stream>>>(
        value, WvP, bv, (const float*)nullptr, 0.f, v2d, kMV, kC, kC);
  }
  // K2: proj = query @ Wpack + bpack (M=32768, N=112) NT=7 -> 2048 waves
  {
    int waves = (kMQ / 16) * (kNPack / 112);
    msda_wmma_gemm_f32<7><<<dim3((waves + 7) / 8), dim3(256), 0, stream>>>(
        query, WpackP, bpack, (const float*)nullptr, 0.f, proj, kMQ, kNPack, kC);
  }
  // K2b: per-head softmax over PTS
  msda_attn_softmax_kernel<<<dim3(kMQ / 256), dim3(256), 0, stream>>>(proj, attn);
  // K3: sampling pass A -> height logits
  msda_height_logits_kernel<<<dim3(kMQ / 8), dim3(256), 0, stream>>>(
      v2d, proj, attn, grd_col, grd_row, u, c0, p);
  // K4a: softmax over NV + argmax
  msda_height_softmax_kernel<<<dim3(kMQ / 256), dim3(256), 0, stream>>>(p, out_idx);
  // K4b: sampling pass B -> acc
  msda_weighted_sample_kernel<<<dim3(kMQ / 8), dim3(256), 0, stream>>>(
      v2d, proj, attn, grd_col, grd_row, p, acc);
  // K5: out = acc @ Wout + bout + 2*query   (M=32768, N=256) NT=4 -> 8192 waves
  {
    int waves = (kMQ / 16) * (kC / 64);
    msda_wmma_gemm_f32<4><<<dim3((waves + 7) / 8), dim3(256), 0, stream>>>(
        acc, WoutP, bout, query, 2.0f, out, kMQ, kC, kC);
  }
}